// SkelConvModel_11158325035624
// MI455X (gfx1250) — compile-verified
//
#include <hip/hip_runtime.h>
#include <math.h>

typedef __attribute__((ext_vector_type(2))) float v2f;
typedef __attribute__((ext_vector_type(8))) float v8f;

#define BN_EPS 1e-5f

__device__ __forceinline__ float silu_f(float x) {
    return x / (1.0f + __expf(-x));
}

// ---------------- pointwise / setup kernels ----------------

__global__ void k_zero(float* p, int n) {
    int i = blockIdx.x * blockDim.x + threadIdx.x;
    if (i < n) p[i] = 0.0f;
}

// COO scatter (indices are unique in this pattern, so plain store == add)
__global__ void k_scatter(float* out, const int* idx, const float* vals, int nnz, int n) {
    int j = blockIdx.x * blockDim.x + threadIdx.x;
    if (j < nnz) {
        int r = idx[j];
        int c = idx[nnz + j];
        out[r * n + c] = vals[j];
    }
}

// Q ((s*s) x (sc*sc)) = kron(Ap (s x sc), Bp (s x sc))
__global__ void k_kron(float* q, const float* Ap, const float* Bp, int s, int sc) {
    int cols = sc * sc;
    int total = s * s * cols;
    int n = blockIdx.x * blockDim.x + threadIdx.x;
    if (n >= total) return;
    int col = n % cols, row = n / cols;
    int i = row / s, r = row % s;
    int j = col / sc, c = col % sc;
    q[n] = Ap[i * sc + j] * Bp[r * sc + c];
}

// copy (rows x cols) into (rows x 16) zero-padded
__global__ void k_pad16(const float* in, float* out, int rows, int cols) {
    int total = rows * 16;
    int n = blockIdx.x * blockDim.x + threadIdx.x;
    if (n >= total) return;
    int c = n % 16, r = n / 16;
    out[n] = (c < cols) ? in[r * cols + c] : 0.0f;
}

// per-grid-row batchnorm stats: x is (64, g, g, F); stats over (b, j, f) per i
__global__ void k_bn_stats(const float* x, float* mean, float* var, int g, int F) {
    int i = blockIdx.x;
    int count = 64 * g * F;
    __shared__ float ssum[256], ssq[256];
    float s = 0.0f, q = 0.0f;
    for (int t = threadIdx.x; t < count; t += 256) {
        int f = t % F;
        int rem = t / F;
        int j = rem % g;
        int b = rem / g;
        float v = x[(((b * g + i) * g) + j) * F + f];
        s += v; q += v * v;
    }
    ssum[threadIdx.x] = s; ssq[threadIdx.x] = q;
    __syncthreads();
    for (int off = 128; off > 0; off >>= 1) {
        if (threadIdx.x < off) {
            ssum[threadIdx.x] += ssum[threadIdx.x + off];
            ssq[threadIdx.x]  += ssq[threadIdx.x + off];
        }
        __syncthreads();
    }
    if (threadIdx.x == 0) {
        float m = ssum[0] / (float)count;
        mean[i] = m;
        var[i]  = ssq[0] / (float)count - m * m;
    }
}

__global__ void k_bn_apply(const float* x, float* z, const float* mean, const float* var,
                           const float* w, const float* bb, int g, int F, int total) {
    int n = blockIdx.x * blockDim.x + threadIdx.x;
    if (n >= total) return;
    int i = (n / (F * g)) % g;
    z[n] = (x[n] - mean[i]) * rsqrtf(var[i] + BN_EPS) * w[i] + bb[i];
}

// (B, R, C) -> (B, C, R)
__global__ void k_transpose(const float* in, float* out, int B, int R, int C) {
    int total = B * R * C;
    int n = blockIdx.x * blockDim.x + threadIdx.x;
    if (n >= total) return;
    int c = n % C, r = (n / C) % R, b = n / (C * R);
    out[(b * C + c) * R + r] = in[n];
}

// out[b,g,f] = act( yg[b,f,g] + yf[b,g,f] ); yf has row stride ldYf (>= F)
__global__ void k_combine(const float* yg, const float* yf, float* out,
                          int B, int G, int F, int ldYf, int act) {
    int total = B * G * F;
    int n = blockIdx.x * blockDim.x + threadIdx.x;
    if (n >= total) return;
    int f = n % F, g = (n / F) % G;
    int b = n / (F * G);
    float v = yg[(b * F + f) * G + g] + yf[(b * G + g) * ldYf + f];
    out[n] = act ? silu_f(v) : v;
}

__global__ void k_silu(const float* in, float* out, int n) {
    int i = blockIdx.x * blockDim.x + threadIdx.x;
    if (i < n) out[i] = silu_f(in[i]);
}

// ---------------- fp32 WMMA GEMMs ----------------
// C(MxN) = act(A(MxK) @ B(KxN)), row-major. REQUIRES M%16==0, K%4==0.
// No bounds checks -> EXEC all-ones throughout (required for v_wmma).
// Running pointers (no per-step k*N multiply); next K-step fragments are
// loaded before the current WMMA issues (software pipelining).

// one 16x16 C tile per wave (used for N%16==0 cases, i.e. N==16 here)
__global__ void __launch_bounds__(256) k_wmma_gemm(const float* __restrict__ A,
                                                   const float* __restrict__ Bm,
                                                   float* __restrict__ C,
                                                   int M, int N, int K, int act) {
    int tilesN = N >> 4;
    int tilesM = M >> 4;
    int w = blockIdx.x * 8 + (threadIdx.x >> 5);          // wave-uniform
    if (w >= tilesM * tilesN) return;
    int tm = w / tilesN, tn = w % tilesN;

    int lane = threadIdx.x & 31;
    int lo = lane & 15;
    int hi = lane >> 4;
    const float* pA  = A  + (size_t)(tm * 16 + lo) * K + 2 * hi;
    const float* pB0 = Bm + (size_t)(2 * hi) * N + tn * 16 + lo;  // row ka
    const float* pB1 = pB0 + N;                                   // row ka+1
    size_t stepB = (size_t)N * 4;

    v8f acc = {};
    v2f a = *(const v2f*)pA;
    v2f b; b.x = pB0[0]; b.y = pB1[0];

    for (int k = 4; k < K; k += 4) {
        pA += 4; pB0 += stepB; pB1 += stepB;
        v2f an = *(const v2f*)pA;
        v2f bn; bn.x = pB0[0]; bn.y = pB1[0];
        acc = __builtin_amdgcn_wmma_f32_16x16x4_f32(
            false, a, false, b, (short)0, acc, false, false);
        a = an; b = bn;
    }
    acc = __builtin_amdgcn_wmma_f32_16x16x4_f32(
        false, a, false, b, (short)0, acc, false, false);

    float* Crow = C + (size_t)(tm * 16 + hi * 8) * N + tn * 16 + lo;
#pragma unroll
    for (int r = 0; r < 8; ++r) {
        float v = acc[r];
        if (act) v = silu_f(v);
        Crow[(size_t)r * N] = v;
    }
}

// one 16x32 C tile per wave: two accumulators share the A fragment.
// REQUIRES additionally N%32==0. Second B tile at +16 cols = immediate
// offset 64B on the same addresses.
__global__ void __launch_bounds__(256) k_wmma_gemm2(const float* __restrict__ A,
                                                    const float* __restrict__ Bm,
                                                    float* __restrict__ C,
                                                    int M, int N, int K, int act) {
    int tilesN = N >> 5;
    int tilesM = M >> 4;
    int w = blockIdx.x * 8 + (threadIdx.x >> 5);          // wave-uniform
    if (w >= tilesM * tilesN) return;
    int tm = w / tilesN, tn = w % tilesN;

    int lane = threadIdx.x & 31;
    int lo = lane & 15;
    int hi = lane >> 4;
    const float* pA  = A  + (size_t)(tm * 16 + lo) * K + 2 * hi;
    const float* pB0 = Bm + (size_t)(2 * hi) * N + tn * 32 + lo;  // row ka
    const float* pB1 = pB0 + N;                                   // row ka+1
    size_t stepB = (size_t)N * 4;

    v8f acc0 = {}, acc1 = {};
    v2f a = *(const v2f*)pA;
    v2f b0, b1;
    b0.x = pB0[0];  b1.x = pB0[16];
    b0.y = pB1[0];  b1.y = pB1[16];

    for (int k = 4; k < K; k += 4) {
        pA += 4; pB0 += stepB; pB1 += stepB;
        v2f an = *(const v2f*)pA;
        v2f bn0, bn1;
        bn0.x = pB0[0];  bn1.x = pB0[16];
        bn0.y = pB1[0];  bn1.y = pB1[16];
        acc0 = __builtin_amdgcn_wmma_f32_16x16x4_f32(
            false, a, false, b0, (short)0, acc0, false, false);
        acc1 = __builtin_amdgcn_wmma_f32_16x16x4_f32(
            false, a, false, b1, (short)0, acc1, false, false);
        a = an; b0 = bn0; b1 = bn1;
    }
    acc0 = __builtin_amdgcn_wmma_f32_16x16x4_f32(
        false, a, false, b0, (short)0, acc0, false, false);
    acc1 = __builtin_amdgcn_wmma_f32_16x16x4_f32(
        false, a, false, b1, (short)0, acc1, false, false);

    float* Crow = C + (size_t)(tm * 16 + hi * 8) * N + tn * 32 + lo;
#pragma unroll
    for (int r = 0; r < 8; ++r) {
        float v0 = acc0[r], v1 = acc1[r];
        if (act) { v0 = silu_f(v0); v1 = silu_f(v1); }
        Crow[(size_t)r * N]      = v0;
        Crow[(size_t)r * N + 16] = v1;
    }
}

// ---------------- host orchestration ----------------

static inline int cdiv(int a, int b) { return (a + b - 1) / b; }

extern "C" void kernel_launch(void* const* d_in, const int* in_sizes, int n_in,
                              void* d_out, int out_size, void* d_ws, size_t ws_size,
                              hipStream_t stream) {
    // Input map (setup_inputs dict order, lists flattened):
    const float* x        = (const float*)d_in[0];   // (64, 8192) = (B, 1024, 8)
    const float* gv64     = (const float*)d_in[1];   // grid_vals g=8
    const float* gv256    = (const float*)d_in[2];   // g=16
    const float* gv1024   = (const float*)d_in[3];   // g=32
    const int*   gi64     = (const int*)d_in[4];     // (2, nnz)
    const int*   gi256    = (const int*)d_in[5];
    const int*   gi1024   = (const int*)d_in[6];
    const float* pA0      = (const float*)d_in[7];   // (16, 8)
    const float* pA1      = (const float*)d_in[8];   // (32, 16)
    const float* pB0      = (const float*)d_in[9];   // (16, 8)
    const float* pB1      = (const float*)d_in[10];  // (32, 16)
    const float* fg0      = (const float*)d_in[11];  // (8, 8)
    const float* fg1      = (const float*)d_in[12];  // (16, 16)
    const float* fg2      = (const float*)d_in[13];  // (32, 32)
    const float* fpm0     = (const float*)d_in[14];  // (8, 16)
    const float* fpm1     = (const float*)d_in[15];  // (16, 32)
    const float* bnw0     = (const float*)d_in[16];  // (32,)
    const float* bnw1     = (const float*)d_in[17];  // (16,)
    const float* bnb0     = (const float*)d_in[18];  // (32,)
    const float* bnb1     = (const float*)d_in[19];  // (16,)
    float* out = (float*)d_out;                      // (64, 2048)
    float* W = (float*)d_ws;

    int nnz64   = in_sizes[1];
    int nnz256  = in_sizes[2];
    int nnz1024 = in_sizes[3];

    // ---- workspace layout (float offsets) ----
    const size_t O_G1024 = 0;              // 1024*1024
    const size_t O_G256  = 1048576;        // 256*256
    const size_t O_G64   = 1114112;        // 64*64
    const size_t O_Q0    = 1118208;        // 1024*256 = kron(pA1,pB1)
    const size_t O_Q1    = 1380352;        // 256*64   = kron(pA0,pB0)
    const size_t O_MEAN  = 1396736;        // 64
    const size_t O_VAR   = 1396800;        // 64
    const size_t O_FG0P  = 1396864;        // 8x16 zero-padded feat_g[0]
    const size_t S       = 1396992;        // scratch pool (3145728 floats)
    // total = 4,542,720 floats = ~18.2 MB

    const int T = 256;

    // ---- build dense grid matrices & kron pooling matrices ----
    k_zero<<<cdiv(1118208, T), T, 0, stream>>>(W + O_G1024, 1118208); // G1024|G256|G64
    k_scatter<<<cdiv(nnz1024, T), T, 0, stream>>>(W + O_G1024, gi1024, gv1024, nnz1024, 1024);
    k_scatter<<<cdiv(nnz256,  T), T, 0, stream>>>(W + O_G256,  gi256,  gv256,  nnz256,  256);
    k_scatter<<<cdiv(nnz64,   T), T, 0, stream>>>(W + O_G64,   gi64,   gv64,   nnz64,   64);
    k_kron<<<cdiv(1024 * 256, T), T, 0, stream>>>(W + O_Q0, pA1, pB1, 32, 16);
    k_kron<<<cdiv(256 * 64,   T), T, 0, stream>>>(W + O_Q1, pA0, pB0, 16, 8);
    k_pad16<<<1, 128, 0, stream>>>(fg0, W + O_FG0P, 8, 8);

    // ================= stage 0: G=1024 (32x32 grid), F=8 -> (256, 16) =================
    float* z0   = W + S;             // (B,1024,8)    524288
    float* z0t  = W + S + 524288;    // (B,8,1024)    524288
    float* Yg0  = W + S + 1048576;   // (B,8,1024)    524288
    float* Yf0p = W + S + 1572864;   // (B,1024,16)  1048576 (padded N=16)
    float* y0   = W + S + 2621440;   // (B,1024,8)    524288
    k_bn_stats<<<32, 256, 0, stream>>>(x, W + O_MEAN, W + O_VAR, 32, 8);
    k_bn_apply<<<cdiv(524288, T), T, 0, stream>>>(x, z0, W + O_MEAN, W + O_VAR,
                                                  bnw0, bnb0, 32, 8, 524288);
    k_transpose<<<cdiv(524288, T), T, 0, stream>>>(z0, z0t, 64, 1024, 8);
    // grid GEMM: (64*8 x 1024) @ (1024 x 1024); 16x32 tiles
    k_wmma_gemm2<<<cdiv(32 * 32, 8), 256, 0, stream>>>(z0t, W + O_G1024, Yg0, 512, 1024, 1024, 0);
    // feature GEMM (padded): (64*1024 x 8) @ (8 x 16)
    k_wmma_gemm<<<cdiv(4096, 8), 256, 0, stream>>>(z0, W + O_FG0P, Yf0p, 65536, 16, 8, 0);
    k_combine<<<cdiv(524288, T), T, 0, stream>>>(Yg0, Yf0p, y0, 64, 1024, 8, 16, 1); // silu
    // pooling: feature then grid
    float* t0   = W + S;             // (B,1024,16) 1048576 (reuses z0+z0t)
    float* t0t  = W + S + 1048576;   // (B,16,1024) 1048576 (reuses Yg0 + low half of Yf0p)
    float* u0   = W + S + 2097152;   // (B,16,256)   262144 (reuses high half of Yf0p)
    float* cur1 = W + S + 2359296;   // (B,256,16)   262144
    k_wmma_gemm<<<cdiv(4096, 8), 256, 0, stream>>>(y0, fpm0, t0, 65536, 16, 8, 0);
    k_transpose<<<cdiv(1048576, T), T, 0, stream>>>(t0, t0t, 64, 1024, 16);
    // (64*16 x 1024) @ (1024 x 256); 16x32 tiles
    k_wmma_gemm2<<<cdiv(64 * 8, 8), 256, 0, stream>>>(t0t, W + O_Q0, u0, 1024, 256, 1024, 0);
    k_transpose<<<cdiv(262144, T), T, 0, stream>>>(u0, cur1, 64, 16, 256);

    // ================= stage 1: G=256 (16x16 grid), F=16 -> (64, 32) =================
    float* z1  = W + S;             // (B,256,16) 262144
    float* z1t = W + S + 262144;    // (B,16,256)
    float* Yg1 = W + S + 524288;    // (B,16,256)
    float* Yf1 = W + S + 786432;    // (B,256,16)
    float* y1  = W + S + 1048576;   // (B,256,16)
    float* t1  = W + S + 1310720;   // (B,256,32) 524288
    float* t1t = W + S + 1835008;   // (B,32,256) 524288 (ends at S+2359296, below cur1)
    k_bn_stats<<<16, 256, 0, stream>>>(cur1, W + O_MEAN, W + O_VAR, 16, 16);
    k_bn_apply<<<cdiv(262144, T), T, 0, stream>>>(cur1, z1, W + O_MEAN, W + O_VAR,
                                                  bnw1, bnb1, 16, 16, 262144);
    k_transpose<<<cdiv(262144, T), T, 0, stream>>>(z1, z1t, 64, 256, 16);
    // (64*16 x 256) @ (256 x 256); 16x32 tiles
    k_wmma_gemm2<<<cdiv(64 * 8, 8), 256, 0, stream>>>(z1t, W + O_G256, Yg1, 1024, 256, 256, 0);
    // (64*256 x 16) @ (16 x 16)
    k_wmma_gemm<<<cdiv(1024, 8), 256, 0, stream>>>(z1, fg1, Yf1, 16384, 16, 16, 0);
    k_combine<<<cdiv(262144, T), T, 0, stream>>>(Yg1, Yf1, y1, 64, 256, 16, 16, 1); // silu
    // (64*256 x 16) @ (16 x 32); 16x32 tiles
    k_wmma_gemm2<<<cdiv(1024, 8), 256, 0, stream>>>(y1, fpm1, t1, 16384, 32, 16, 0);
    k_transpose<<<cdiv(524288, T), T, 0, stream>>>(t1, t1t, 64, 256, 32);
    float* u1   = W + S;            // (B,32,64) 131072
    float* cur2 = W + S + 131072;   // (B,64,32) 131072
    // (64*32 x 256) @ (256 x 64); 16x32 tiles
    k_wmma_gemm2<<<cdiv(128 * 2, 8), 256, 0, stream>>>(t1t, W + O_Q1, u1, 2048, 64, 256, 0);
    k_transpose<<<cdiv(131072, T), T, 0, stream>>>(u1, cur2, 64, 32, 64);

    // ================= final: silu then diag_block(2): G=64, F=32 =================
    float* s2  = W + S + 262144;    // (B,64,32)
    float* s2t = W + S + 393216;    // (B,32,64)
    float* Yg2 = W + S + 524288;    // (B,32,64)
    float* Yf2 = W + S + 655360;    // (B,64,32)
    k_silu<<<cdiv(131072, T), T, 0, stream>>>(cur2, s2, 131072);
    k_transpose<<<cdiv(131072, T), T, 0, stream>>>(s2, s2t, 64, 64, 32);
    // (64*32 x 64) @ (64 x 64); 16x32 tiles
    k_wmma_gemm2<<<cdiv(128 * 2, 8), 256, 0, stream>>>(s2t, W + O_G64, Yg2, 2048, 64, 64, 0);
    // (64*64 x 32) @ (32 x 32); 16x32 tiles
    k_wmma_gemm2<<<cdiv(256, 8), 256, 0, stream>>>(s2, fg2, Yf2, 4096, 32, 32, 0);
    k_combine<<<cdiv(131072, T), T, 0, stream>>>(Yg2, Yf2, out, 64, 64, 32, 32, 0); // no act
}